// SoftMeritBasedLoss_28965259444639
// MI455X (gfx1250) — compile-verified
//
#include <hip/hip_runtime.h>
#include <hip/hip_bf16.h>
#include <stdint.h>

// ---------------------------------------------------------------------------
// SoftMeritBasedLoss for MI455X (gfx1250), T == 1.0 (fixed init kwarg, so all
// /T scalings are identity and one softmax per matrix suffices).
//
// Streaming single-pass online-softmax reduction:
//   per row: m_l,Z_l over logits; m_t,Z_t,A=sum(w*tg),B=sum(w*l) over targets
//   kld  = A/Z_t - logZt - (B/Z_t - logZl)
//   ce   = logZl - logits[t]
//   alpha= exp(targets[t] - logZt)
//   loss = mean((1-alpha)*ce + alpha*kld)
//
// Data path: per-wave depth-4 LDS ring fed by GLOBAL_LOAD_ASYNC_TO_LDS_B128
// (CDNA5 async-tensor path, ASYNCcnt), consumed via ds_read_b128.
// ---------------------------------------------------------------------------

#define VOCAB    32000
#define WAVES    10
#define BLOCK    (WAVES * 32)            // 320 threads = 10 wave32
#define F4_ROW   (VOCAB / 4)             // 8000 float4 per row per matrix
#define F4_WAVE  (F4_ROW / WAVES)        // 800 float4 per wave
#define NSTEP    (F4_WAVE / 32)          // 25 steps (one float4 per lane/step)
#define DEPTH    4                       // async prefetch depth (ring slots)

// One async b128 per lane: copies 16 bytes from global address sbase + goff to LDS address ldsoff
#define ASYNC_B128(ldsoff, goff, sbase)                                   \
  asm volatile("global_load_async_to_lds_b128 %0, %1, %2"                 \
               :: "v"(ldsoff), "v"(goff), "s"(sbase) : "memory")

#define WAIT_ASYNC(n) asm volatile("s_wait_asynccnt " #n ::: "memory")
#define WAIT_DS0()    asm volatile("s_wait_dscnt 0" ::: "memory")

__device__ __forceinline__ void upd(float l, float tg,
                                    float& m_l, float& Z_l,
                                    float& m_t, float& Z_t,
                                    float& A, float& B) {
  if (l > m_l) { Z_l *= __expf(m_l - l); m_l = l; }
  Z_l += __expf(l - m_l);
  if (tg > m_t) {
    float sc = __expf(m_t - tg);
    Z_t *= sc; A *= sc; B *= sc; m_t = tg;
  }
  float w = __expf(tg - m_t);
  Z_t += w;
  A += w * tg;
  B += w * l;
}

__device__ __forceinline__ void comb2(float& m, float& Z, float om, float oZ) {
  float nm = fmaxf(m, om);
  Z = Z * __expf(m - nm) + oZ * __expf(om - nm);
  m = nm;
}

__device__ __forceinline__ void comb4(float& m, float& Z, float& A, float& B,
                                      float om, float oZ, float oA, float oB) {
  float nm = fmaxf(m, om);
  float s0 = __expf(m - nm);
  float s1 = __expf(om - nm);
  Z = Z * s0 + oZ * s1;
  A = A * s0 + oA * s1;
  B = B * s0 + oB * s1;
  m = nm;
}

__global__ __launch_bounds__(BLOCK)
void smbl_row_kernel(const float* __restrict__ logits,
                     const float* __restrict__ targets,
                     const long long* __restrict__ t,
                     float* __restrict__ row_loss) {
  // [wave][slot][matrix(L/T)][lane] float4 ring = 40 KB, + reduce scratch
  __shared__ float4 ring[WAVES * DEPTH * 2 * 32];
  __shared__ float  red[WAVES][8];

  const int row  = blockIdx.x;
  const int tid  = threadIdx.x;
  const int w    = tid >> 5;
  const int lane = tid & 31;

  const float* rl = logits  + (size_t)row * VOCAB;
  const float* rt = targets + (size_t)row * VOCAB;

  // Thread 0 gathers the target-index scalars early (latency hidden).
  float l_at_t = 0.f, tg_at_t = 0.f;
  if (tid == 0) {
    long long ti = t[row];
    l_at_t  = rl[ti];
    tg_at_t = rt[ti];
  }

  // LDS byte offsets (group-segment relative: low 32 bits of generic ptr).
  const uint32_t lds0     = (uint32_t)(uint64_t)(uintptr_t)(void*)ring;
  const uint32_t lds_lane = lds0 + (uint32_t)w * (DEPTH * 1024u)
                                 + (uint32_t)lane * 16u;
  const uint32_t gofs0 = ((uint32_t)w * F4_WAVE + (uint32_t)lane) * 16u;

  // Prologue: fill the ring (DEPTH steps x 2 matrices = 8 async issues).
#pragma unroll
  for (int st = 0; st < DEPTH; ++st) {
    uint32_t go = gofs0 + (uint32_t)st * 512u;
    uint32_t lo = lds_lane + (uint32_t)st * 1024u;
    ASYNC_B128(lo, go, rl);
    ASYNC_B128(lo + 512u, go, rt);
  }

  float m_l = -INFINITY, Z_l = 0.f;
  float m_t = -INFINITY, Z_t = 0.f, A = 0.f, B = 0.f;

  const int base_f4 = (w * DEPTH * 2) * 32 + lane;

  // Main loop: wait oldest step, consume, refill its slot with step s+DEPTH.
  for (int s = 0; s < NSTEP - DEPTH; ++s) {
    WAIT_ASYNC(6);                       // 2*(DEPTH-1): step s retired
    const int slot = s & (DEPTH - 1);
    float4 Lv = ring[base_f4 + slot * 64];
    float4 Tv = ring[base_f4 + slot * 64 + 32];
    WAIT_DS0();                          // slot data is in VGPRs -> reusable
    uint32_t go = gofs0 + (uint32_t)(s + DEPTH) * 512u;
    uint32_t lo = lds_lane + (uint32_t)slot * 1024u;
    ASYNC_B128(lo, go, rl);
    ASYNC_B128(lo + 512u, go, rt);
    upd(Lv.x, Tv.x, m_l, Z_l, m_t, Z_t, A, B);
    upd(Lv.y, Tv.y, m_l, Z_l, m_t, Z_t, A, B);
    upd(Lv.z, Tv.z, m_l, Z_l, m_t, Z_t, A, B);
    upd(Lv.w, Tv.w, m_l, Z_l, m_t, Z_t, A, B);
  }

  // Tail: drain and consume the last DEPTH steps.
  WAIT_ASYNC(0);
#pragma unroll
  for (int s = NSTEP - DEPTH; s < NSTEP; ++s) {
    const int slot = s & (DEPTH - 1);
    float4 Lv = ring[base_f4 + slot * 64];
    float4 Tv = ring[base_f4 + slot * 64 + 32];
    upd(Lv.x, Tv.x, m_l, Z_l, m_t, Z_t, A, B);
    upd(Lv.y, Tv.y, m_l, Z_l, m_t, Z_t, A, B);
    upd(Lv.z, Tv.z, m_l, Z_l, m_t, Z_t, A, B);
    upd(Lv.w, Tv.w, m_l, Z_l, m_t, Z_t, A, B);
  }

  // Wave32 tree reduce.
#pragma unroll
  for (int off = 16; off > 0; off >>= 1) {
    float om  = __shfl_xor(m_l, off, 32);
    float oZ  = __shfl_xor(Z_l, off, 32);
    float omt = __shfl_xor(m_t, off, 32);
    float oZt = __shfl_xor(Z_t, off, 32);
    float oA  = __shfl_xor(A,   off, 32);
    float oB  = __shfl_xor(B,   off, 32);
    comb2(m_l, Z_l, om, oZ);
    comb4(m_t, Z_t, A, B, omt, oZt, oA, oB);
  }

  if (lane == 0) {
    red[w][0] = m_l; red[w][1] = Z_l;
    red[w][2] = m_t; red[w][3] = Z_t;
    red[w][4] = A;   red[w][5] = B;
  }
  __syncthreads();

  if (tid == 0) {
    float fm  = red[0][0], fZ  = red[0][1];
    float fmt = red[0][2], fZt = red[0][3];
    float fA  = red[0][4], fB  = red[0][5];
#pragma unroll
    for (int i = 1; i < WAVES; ++i) {
      comb2(fm, fZ, red[i][0], red[i][1]);
      comb4(fmt, fZt, fA, fB, red[i][2], red[i][3], red[i][4], red[i][5]);
    }
    float logZl = fm  + __logf(fZ);    // full log-partition of logits row
    float logZt = fmt + __logf(fZt);   // full log-partition of targets row
    float invZt = 1.0f / fZt;
    float sum_p_logp = fA * invZt - logZt;   // sum p*log p
    float sum_p_logq = fB * invZt - logZl;   // sum p*log q
    float kld   = sum_p_logp - sum_p_logq;
    float ce    = logZl - l_at_t;            // -log_softmax(logits)[t]
    float alpha = __expf(tg_at_t - logZt);   // softmax(targets)[t]
    row_loss[row] = (1.0f - alpha) * ce + alpha * kld;  // T*T == 1
  }
}

__global__ __launch_bounds__(256)
void smbl_finalize_kernel(const float* __restrict__ row_loss,
                          float* __restrict__ out, int n) {
  __shared__ float sm[256];
  float acc = 0.f;
  for (int i = threadIdx.x; i < n; i += 256) acc += row_loss[i];
  sm[threadIdx.x] = acc;
  __syncthreads();
#pragma unroll
  for (int s = 128; s > 0; s >>= 1) {
    if (threadIdx.x < s) sm[threadIdx.x] += sm[threadIdx.x + s];
    __syncthreads();
  }
  if (threadIdx.x == 0) out[0] = sm[0] / (float)n;
}

extern "C" void kernel_launch(void* const* d_in, const int* in_sizes, int n_in,
                              void* d_out, int out_size, void* d_ws, size_t ws_size,
                              hipStream_t stream) {
  const float*     logits  = (const float*)d_in[0];
  const float*     targets = (const float*)d_in[1];
  const long long* t       = (const long long*)d_in[2];
  float* out = (float*)d_out;
  float* ws  = (float*)d_ws;           // 4096 per-row losses

  const int batch = in_sizes[2];       // 4096

  smbl_row_kernel<<<batch, BLOCK, 0, stream>>>(logits, targets, t, ws);
  smbl_finalize_kernel<<<1, 256, 0, stream>>>(ws, out, batch);
}